// PostDetect_18983755448811
// MI455X (gfx1250) — compile-verified
//
#include <hip/hip_runtime.h>
#include <hip/hip_bf16.h>

// ---------------------------------------------------------------------------
// PostDetect (YOLO head): DFL box decode + sigmoid scores for MI455X (gfx1250)
//   inputs : feat0 (b,144,80,80) f32, feat1 (b,144,40,40) f32, feat2 (b,144,20,20) f32
//   outputs: boxes (b,8400,4) f32  ++  scores (b,8400,80) f32   (flat-concat)
// Memory-bound (~490MB @ 23.3TB/s => ~21us floor). DFL softmax-expectation is
// done with chained V_WMMA_F32_16X16X4_F32 (f32-exact), 16 anchors per wave.
// ---------------------------------------------------------------------------

typedef __attribute__((ext_vector_type(2))) float v2f;
typedef __attribute__((ext_vector_type(8))) float v8f;

#define A_TOTAL 8400
#define NCH     144
#define NCLS    80

// ---------------------------------------------------------------------------
// Kernel 1: DFL box decode. One wave32 handles 16 anchors.
// Rows M = 16 anchors (fixed side), K = 16 bins via 4x V_WMMA_F32_16X16X4_F32.
// A layout (ISA 16x4 f32): lanes 0-15 hold K={0,1}, lanes 16-31 hold K={2,3};
// we load channels in exactly that pattern so WMMA needs no data shuffles.
// B: col0 = bin index, col1 = 1 -> D[:,0]=sum(e*k), D[:,1]=sum(e).
// ---------------------------------------------------------------------------
__global__ __launch_bounds__(256) void boxes_kernel(
    const float* __restrict__ f0, const float* __restrict__ f1,
    const float* __restrict__ f2, float* __restrict__ out_boxes)
{
    __shared__ float sW[8][4][16];
    __shared__ float sT[8][4][16];

    const int tid  = threadIdx.x;
    const int w    = tid >> 5;      // wave in block (0..7)
    const int lane = tid & 31;
    const int bi   = blockIdx.y;

    const int a0    = (blockIdx.x * 8 + w) * 16;   // wave-uniform anchor base
    const bool valid = (a0 < A_TOTAL);
    const int a0c   = valid ? a0 : (A_TOTAL - 16); // clamp: EXEC must stay all-1s

    // Per-wave level (6400/1600/400 all divisible by 16 -> uniform per wave)
    int off, hw, Wd, st;
    const float* fp;
    if (a0c < 6400)      { off = 0;    hw = 6400; Wd = 80; st = 8;  fp = f0; }
    else if (a0c < 8000) { off = 6400; hw = 1600; Wd = 40; st = 16; fp = f1; }
    else                 { off = 8000; hw = 400;  Wd = 20; st = 32; fp = f2; }

    const int  li   = (a0c - off) + (lane & 15);          // anchor within level
    const int  csel = (lane >> 4) & 1;                    // 0: K{0,1}, 1: K{2,3}
    const int  n    = lane & 15;                          // WMMA column for B
    const float* base = fp + (size_t)bi * NCH * hw + li;

    #pragma unroll
    for (int sidx = 0; sidx < 4; ++sidx) {
        // ---- load this lane's 8 of the 16x16 (anchors x bins) exp-matrix ----
        float x[8];
        #pragma unroll
        for (int ch = 0; ch < 4; ++ch) {
            const int c0 = sidx * 16 + ch * 4 + csel * 2;
            x[2 * ch]     = base[(size_t)c0 * hw];
            x[2 * ch + 1] = base[(size_t)(c0 + 1) * hw];
        }
        // ---- row max over 16 bins (8 local + partner lane via xor16) ----
        float mx = x[0];
        #pragma unroll
        for (int i = 1; i < 8; ++i) mx = fmaxf(mx, x[i]);
        mx = fmaxf(mx, __shfl_xor(mx, 16));
        float e[8];
        #pragma unroll
        for (int i = 0; i < 8; ++i) e[i] = __expf(x[i] - mx);

        // ---- 4 chained f32 WMMAs: D[:,0]=sum(e*k), D[:,1]=sum(e) ----
        v8f c8 = {0.f, 0.f, 0.f, 0.f, 0.f, 0.f, 0.f, 0.f};
        #pragma unroll
        for (int ch = 0; ch < 4; ++ch) {
            v2f av; av.x = e[2 * ch]; av.y = e[2 * ch + 1];
            const float k0 = (float)(ch * 4 + csel * 2);
            v2f bv;
            bv.x = (n == 0) ? k0          : ((n == 1) ? 1.0f : 0.0f);
            bv.y = (n == 0) ? (k0 + 1.0f) : ((n == 1) ? 1.0f : 0.0f);
            c8 = __builtin_amdgcn_wmma_f32_16x16x4_f32(
                    false, av, false, bv, (short)0, c8, false, false);
        }

        // ---- export the two useful D columns to LDS (per-wave region) ----
        if (lane == 0) {
            #pragma unroll
            for (int j = 0; j < 8; ++j) sW[w][sidx][j] = c8[j];        // M=0..7 wsum
        } else if (lane == 1) {
            #pragma unroll
            for (int j = 0; j < 8; ++j) sT[w][sidx][j] = c8[j];        // M=0..7 tsum
        } else if (lane == 16) {
            #pragma unroll
            for (int j = 0; j < 8; ++j) sW[w][sidx][8 + j] = c8[j];    // M=8..15 wsum
        } else if (lane == 17) {
            #pragma unroll
            for (int j = 0; j < 8; ++j) sT[w][sidx][8 + j] = c8[j];    // M=8..15 tsum
        }
    }

    __syncthreads();  // all waves participate (clamped waves did full work)

    if (valid && lane < 16) {
        const float d0 = sW[w][0][lane] / sT[w][0][lane];
        const float d1 = sW[w][1][lane] / sT[w][1][lane];
        const float d2 = sW[w][2][lane] / sT[w][2][lane];
        const float d3 = sW[w][3][lane] / sT[w][3][lane];

        const int loc = (a0c - off) + lane;
        const float gx = (float)(loc % Wd) + 0.5f;
        const float gy = (float)(loc / Wd) + 0.5f;
        const float fs = (float)st;

        float4 r;
        r.x = (gx - d0) * fs;
        r.y = (gy - d1) * fs;
        r.z = (gx + d2) * fs;
        r.w = (gy + d3) * fs;
        reinterpret_cast<float4*>(out_boxes)[(size_t)bi * A_TOTAL + a0c + lane] = r;
    }
}

// ---------------------------------------------------------------------------
// Kernel 2: sigmoid scores with LDS tile transpose (channel-major in ->
// anchor-major out). Block = 64 anchors x 80 channels; reads coalesced along
// anchors, writes coalesced along the flat (a,c) output. Level uniform per
// block (level offsets 0/6400/8000 all divisible by 64).
// ---------------------------------------------------------------------------
__global__ __launch_bounds__(256) void scores_kernel(
    const float* __restrict__ f0, const float* __restrict__ f1,
    const float* __restrict__ f2, float* __restrict__ out_scores)
{
    __shared__ float tile[NCLS][65];   // +1 pad: conflict-free transpose

    const int tid = threadIdx.x;
    const int bi  = blockIdx.y;
    const int a0  = blockIdx.x * 64;

    int off, hw;
    const float* fp;
    if (a0 < 6400)      { off = 0;    hw = 6400; fp = f0; }
    else if (a0 < 8000) { off = 6400; hw = 1600; fp = f1; }
    else                { off = 8000; hw = 400;  fp = f2; }

    const int na = min(64, A_TOTAL - a0);           // anchors in this tile
    const size_t ibase = (size_t)bi * NCH * hw + (a0 - off);

    // read phase: 4 channels x 64 anchors per step, 20 steps
    const int a = tid & 63;
    if (a < na) {
        #pragma unroll 4
        for (int c = tid >> 6; c < NCLS; c += 4)
            tile[c][a] = fp[ibase + (size_t)(64 + c) * hw + a];
    }
    __syncthreads();

    // write phase: flat (a,c) -> fully coalesced global stores
    const size_t obase = ((size_t)bi * A_TOTAL + a0) * NCLS;
    const int nel = na * NCLS;
    for (int i = tid; i < nel; i += 256) {
        const int aa = i / NCLS;
        const int cc = i - aa * NCLS;
        const float x = tile[cc][aa];
        out_scores[obase + i] = 1.0f / (1.0f + __expf(-x));
    }
}

// ---------------------------------------------------------------------------
extern "C" void kernel_launch(void* const* d_in, const int* in_sizes, int n_in,
                              void* d_out, int out_size, void* d_ws, size_t ws_size,
                              hipStream_t stream)
{
    const float* f0 = (const float*)d_in[0];
    const float* f1 = (const float*)d_in[1];
    const float* f2 = (const float*)d_in[2];

    const int b = in_sizes[0] / (NCH * 6400);       // batch from flat size

    float* boxes  = (float*)d_out;
    float* scores = boxes + (size_t)b * A_TOTAL * 4;

    // boxes: 16 anchors/wave, 8 waves/block -> 128 anchors/block
    dim3 gb((A_TOTAL + 127) / 128, b);
    boxes_kernel<<<gb, dim3(256), 0, stream>>>(f0, f1, f2, boxes);

    // scores: 64 anchors/block
    dim3 gs((A_TOTAL + 63) / 64, b);
    scores_kernel<<<gs, dim3(256), 0, stream>>>(f0, f1, f2, scores);
}